// MultiInstanceAlign_25915832664650
// MI455X (gfx1250) — compile-verified
//
#include <hip/hip_runtime.h>
#include <hip/hip_bf16.h>
#include <stdint.h>

#define PHh 8
#define PWw 32
#define NINST 4
#define CCH 256
#define CCHUNK 32
#define NPIX (PHh * PWw)   // 256 output pixels per (roi, channel)

struct Tap { int lo, hi; float fr, val; };

__device__ __forceinline__ Tap make_tap(float c, int size) {
    Tap t;
    const float fs = (float)size;
    const bool valid = (c >= -1.0f) && (c <= fs);
    c = fmaxf(c, 0.0f);
    int lo = (int)floorf(c);
    const bool atedge = (lo >= size - 1);
    lo = min(lo, size - 1);
    t.hi  = min(lo + 1, size - 1);
    t.lo  = lo;
    t.fr  = atedge ? 0.0f : (c - (float)lo);
    t.val = valid ? 1.0f : 0.0f;
    return t;
}

__global__ __launch_bounds__(256) void mia_kernel(
    const float* __restrict__ f0, const float* __restrict__ f1,
    const float* __restrict__ f2, const float* __restrict__ f3,
    const float* __restrict__ rois, float* __restrict__ out)
{
    __shared__ float buf[2][NPIX * 16];   // 32 KB: 16 gathered taps per thread, double-buffered

    const int n   = blockIdx.x;
    const int c0  = blockIdx.y * CCHUNK;
    const int tid = threadIdx.x;
    const int ph  = tid >> 5;
    const int pw  = tid & 31;

    // ---- roi + instance boxes (uniform per block; cheap redundant compute) ----
    const float* R = rois + (size_t)n * (NINST * 5);
    const float bidf = R[0];
    float ix1[NINST], iy1[NINST], ix2[NINST], iy2[NINST];
    float bx1 = 1e30f, by1 = 1e30f, bx2 = -1e30f, by2 = -1e30f;
#pragma unroll
    for (int i = 0; i < NINST; ++i) {
        ix1[i] = R[i * 5 + 1]; iy1[i] = R[i * 5 + 2];
        ix2[i] = R[i * 5 + 3]; iy2[i] = R[i * 5 + 4];
        bx1 = fminf(bx1, ix1[i]); by1 = fminf(by1, iy1[i]);
        bx2 = fmaxf(bx2, ix2[i]); by2 = fmaxf(by2, iy2[i]);
    }

    // ---- FPN level select ----
    const float s = sqrtf((by2 - by1) * (bx2 - bx1));
    int lvl = (int)floorf(4.0f + log2f(s * (1.0f / 224.0f) + 1e-6f));
    lvl = min(max(lvl, 2), 5);
    const int level = lvl - 2;
    const float* feat; int H, W; float scale;
    if      (level == 0) { feat = f0; H = 256; W = 256; scale = 0.25f;    }
    else if (level == 1) { feat = f1; H = 128; W = 128; scale = 0.125f;   }
    else if (level == 2) { feat = f2; H = 64;  W = 64;  scale = 0.0625f;  }
    else                 { feat = f3; H = 32;  W = 32;  scale = 0.03125f; }

    // ---- per-thread sampling taps (channel-invariant) ----
    const float x1 = bx1 * scale, y1 = by1 * scale;
    const float rw = fmaxf(bx2 * scale - x1, 1.0f);
    const float rh = fmaxf(by2 * scale - y1, 1.0f);
    const float sxs = rw * (1.0f / 64.0f);   // PW*SR
    const float sys = rh * (1.0f / 16.0f);   // PH*SR
    const Tap TY[2] = { make_tap(y1 + ((float)(2 * ph) + 0.5f) * sys, H),
                        make_tap(y1 + ((float)(2 * ph) + 1.5f) * sys, H) };
    const Tap TX[2] = { make_tap(x1 + ((float)(2 * pw) + 0.5f) * sxs, W),
                        make_tap(x1 + ((float)(2 * pw) + 1.5f) * sxs, W) };

    int   off[16]; float wt[16];
    {
        int k = 0;
#pragma unroll
        for (int a = 0; a < 2; ++a) {
#pragma unroll
            for (int b = 0; b < 2; ++b) {
                const Tap Y = TY[a], X = TX[b];
                const float vv = 0.25f * Y.val * X.val;  // 2x2 mean + sample validity
                const float fy = Y.fr, fx = X.fr;
                off[k] = Y.lo * W + X.lo; wt[k] = vv * (1.0f - fy) * (1.0f - fx); ++k;
                off[k] = Y.lo * W + X.hi; wt[k] = vv * (1.0f - fy) * fx;          ++k;
                off[k] = Y.hi * W + X.lo; wt[k] = vv * fy * (1.0f - fx);          ++k;
                off[k] = Y.hi * W + X.hi; wt[k] = vv * fy * fx;                   ++k;
            }
        }
    }

    // ---- per-instance masks at this (ph, pw) ----
    const float wr = 32.0f / (bx2 - bx1);
    const float hr =  8.0f / (by2 - by1);
    float m[NINST];
#pragma unroll
    for (int i = 0; i < NINST; ++i) {
        const int px1 = (int)((ix1[i] - bx1) * wr);
        const int py1 = (int)((iy1[i] - by1) * hr);
        const int px2 = (int)((ix2[i] - bx1) * wr);
        const int py2 = (int)((iy2[i] - by1) * hr);
        m[i] = (ph >= py1 && ph < py2 && pw >= px1 && pw < px2) ? 1.0f : 0.0f;
    }

    // ---- channel loop: double-buffered async gather-to-LDS pipeline ----
    const int bid = (int)bidf;
    const size_t plane = (size_t)H * (size_t)W;
    const float* base = feat + ((size_t)bid * CCH + (size_t)c0) * plane;
    const uint32_t l0 = (uint32_t)(uintptr_t)&buf[0][tid * 16];
    const uint32_t l1 = (uint32_t)(uintptr_t)&buf[1][tid * 16];

    auto issue = [&](int cc, uint32_t laddr) {
        const float* p = base + (size_t)cc * plane;
#pragma unroll
        for (int k = 0; k < 16; ++k) {
            const float* g = p + off[k];
            const uint32_t l = laddr + (uint32_t)(k * 4);
            // per-lane gather from global directly into this lane's LDS slot (ASYNCcnt)
            asm volatile("global_load_async_to_lds_b32 %0, %1, off"
                         :: "v"(l), "v"(g) : "memory");
        }
    };

    issue(0, l0);
    const size_t obase = ((size_t)n * (NINST * CCH) + (size_t)c0) * NPIX + (size_t)tid;
    const size_t istr  = (size_t)CCH * NPIX;   // instance stride in output

#pragma unroll 1
    for (int cc = 0; cc < CCHUNK; ++cc) {
        const int b = cc & 1;
        if (cc + 1 < CCHUNK) {
            issue(cc + 1, (b ^ 1) ? l1 : l0);
            asm volatile("s_wait_asynccnt 16" ::: "memory");  // drain channel cc's 16 ops
        } else {
            asm volatile("s_wait_asynccnt 0" ::: "memory");
        }
        const float* lb = &buf[b][tid * 16];
        float acc = 0.0f;
#pragma unroll
        for (int k = 0; k < 16; ++k) acc += wt[k] * lb[k];

        const size_t o = obase + (size_t)cc * NPIX;
        out[o]            = acc * m[0];
        out[o + istr]     = acc * m[1];
        out[o + 2 * istr] = acc * m[2];
        out[o + 3 * istr] = acc * m[3];
    }
}

extern "C" void kernel_launch(void* const* d_in, const int* in_sizes, int n_in,
                              void* d_out, int out_size, void* d_ws, size_t ws_size,
                              hipStream_t stream) {
    const float* f0   = (const float*)d_in[0];
    const float* f1   = (const float*)d_in[1];
    const float* f2   = (const float*)d_in[2];
    const float* f3   = (const float*)d_in[3];
    const float* rois = (const float*)d_in[4];
    float* out = (float*)d_out;

    const int N = in_sizes[4] / (NINST * 5);   // 256 rois
    dim3 grid(N, CCH / CCHUNK);                // (256, 8)
    mia_kernel<<<grid, 256, 0, stream>>>(f0, f1, f2, f3, rois, out);
}